// ConvertMomentsTEtoParamsTEGaussLayer_50002009260129
// MI455X (gfx1250) — compile-verified
//
#include <hip/hip_runtime.h>

typedef __attribute__((ext_vector_type(2))) float v2f;
typedef __attribute__((ext_vector_type(8))) float v8f;

#define NN 128      // matrix dim
#define LS 132      // padded LDS row stride (floats); 132 % 64 = 4 -> spreads banks

// contiguous 8B LDS load -> ds_load_b64 into an aligned VGPR pair
__device__ __forceinline__ v2f ld2(const float* p) {
    return *reinterpret_cast<const v2f*>(p);
}

__global__ __launch_bounds__(256) void gauss_te_kernel(
    const float* __restrict__ muTE,
    const float* __restrict__ covTE,
    const float* __restrict__ chol,
    float* __restrict__ out,
    int nbatch)
{
    const int batch = blockIdx.x;
    const int tid   = threadIdx.x;
    const int lane  = tid & 31;
    const int wave  = tid >> 5;          // 0..7: tile-row owned by this wave
    const int n     = lane & 15;         // col within tile (B/C/D layout)
    const int half  = lane >> 4;         // 0 or 1
    const int koff  = half << 1;         // K sub-offset for A/B frags (0 or 2)
    const int rb    = wave << 4;         // row-band base (16*wave)

    extern __shared__ float smem[];
    float* sC = smem;                    // chol, 128 x LS
    float* sT = smem + NN * LS;          // covTE -> T -> M (reused)

    // ---- muTE passthrough ----
    if (tid < NN) out[(size_t)batch * NN + tid] = muTE[(size_t)batch * NN + tid];

    // ---- cooperative float4 load of chol & covTE into LDS ----
    const float4* g4C = reinterpret_cast<const float4*>(chol  + (size_t)batch * NN * NN);
    const float4* g4V = reinterpret_cast<const float4*>(covTE + (size_t)batch * NN * NN);
    #pragma unroll
    for (int i = 0; i < 16; ++i) {
        int idx = tid + i * 256;               // 0..4095 float4s
        int r   = idx >> 5;                    // 32 float4 per row
        int c4  = (idx & 31) << 2;
        float4 vc = g4C[idx];
        float4 vv = g4V[idx];
        *reinterpret_cast<float4*>(&sC[r * LS + c4]) = vc;
        *reinterpret_cast<float4*>(&sT[r * LS + c4]) = vv;
    }
    __syncthreads();

    const v8f vzero = {0.f, 0.f, 0.f, 0.f, 0.f, 0.f, 0.f, 0.f};
    v8f acc[8];

    // ================= phase 1: T = Cov @ C  (wave w -> T row-band w) ========
    #pragma unroll
    for (int jt = 0; jt < 8; ++jt) acc[jt] = vzero;

    for (int kc = 0; kc < 32; ++kc) {
        const int k0 = kc << 2;
        // A = Cov rows (16x4): two adjacent floats in one row -> ds_load_b64
        v2f a = ld2(&sT[(rb + n) * LS + k0 + koff]);
        #pragma unroll
        for (int jt = 0; jt < 8; ++jt) {
            v2f b;                                     // B = C (4x16): 2 rows
            b.x = sC[(k0 + koff)     * LS + (jt << 4) + n];
            b.y = sC[(k0 + koff + 1) * LS + (jt << 4) + n];
            acc[jt] = __builtin_amdgcn_wmma_f32_16x16x4_f32(
                false, a, false, b, (short)0, acc[jt], false, false);
        }
    }
    // overwrite own Cov row-band with T (only this wave ever touched it)
    #pragma unroll
    for (int jt = 0; jt < 8; ++jt)
        #pragma unroll
        for (int r = 0; r < 8; ++r)
            sT[(rb + r + 8 * half) * LS + (jt << 4) + n] = acc[jt][r];
    __syncthreads();

    // ================= phase 2: phi = C^T @ T (only tiles jt <= wave) ========
    #pragma unroll
    for (int jt = 0; jt < 8; ++jt) acc[jt] = vzero;

    for (int kc = 0; kc < 32; ++kc) {
        const int k0 = kc << 2;
        v2f a;                   // A = C^T -> C column band (2 rows, 1 col)
        a.x = sC[(k0 + koff)     * LS + rb + n];
        a.y = sC[(k0 + koff + 1) * LS + rb + n];
        for (int jt = 0; jt <= wave; ++jt) {           // triangular skip
            v2f b;                                     // B = T (4x16): 2 rows
            b.x = sT[(k0 + koff)     * LS + (jt << 4) + n];
            b.y = sT[(k0 + koff + 1) * LS + (jt << 4) + n];
            acc[jt] = __builtin_amdgcn_wmma_f32_16x16x4_f32(
                false, a, false, b, (short)0, acc[jt], false, false);
        }
    }
    __syncthreads();   // everyone done reading T

    // ---- M = mask * phi, written over T buffer (own row-band only) ----
    for (int jt = wave + 1; jt < 8; ++jt)              // strictly above diag: 0
        #pragma unroll
        for (int r = 0; r < 8; ++r)
            sT[(rb + r + 8 * half) * LS + (jt << 4) + n] = 0.0f;
    for (int jt = 0; jt < wave; ++jt)                  // strictly below diag: 1
        #pragma unroll
        for (int r = 0; r < 8; ++r)
            sT[(rb + r + 8 * half) * LS + (jt << 4) + n] = acc[jt][r];
    {                                                  // diagonal tile
        const int jt = wave;
        #pragma unroll
        for (int r = 0; r < 8; ++r) {
            int iloc = r + 8 * half;
            float f = (iloc > n) ? 1.0f : ((iloc == n) ? 0.5f : 0.0f);
            sT[(rb + iloc) * LS + (jt << 4) + n] = f * acc[jt][r];
        }
    }
    __syncthreads();

    // ================= phase 3: out = -C @ M  (M rows < 16*jt are zero) ======
    #pragma unroll
    for (int jt = 0; jt < 8; ++jt) acc[jt] = vzero;

    for (int kc = 0; kc < 32; ++kc) {
        const int k0 = kc << 2;
        // A = C rows: adjacent floats -> ds_load_b64
        v2f a = ld2(&sC[(rb + n) * LS + k0 + koff]);
        const int jmax = kc >> 2;                      // tile jt needs kc >= 4*jt
        for (int jt = 0; jt <= jmax; ++jt) {
            v2f b;                                     // B = M (4x16): 2 rows
            b.x = sT[(k0 + koff)     * LS + (jt << 4) + n];
            b.y = sT[(k0 + koff + 1) * LS + (jt << 4) + n];
            acc[jt] = __builtin_amdgcn_wmma_f32_16x16x4_f32(
                false, a, false, b, (short)0, acc[jt], false, false);
        }
    }

    // ---- store cholTE = -result ----
    float* oc = out + (size_t)nbatch * NN + (size_t)batch * NN * NN;
    #pragma unroll
    for (int jt = 0; jt < 8; ++jt)
        #pragma unroll
        for (int r = 0; r < 8; ++r)
            oc[(size_t)(rb + r + 8 * half) * NN + (jt << 4) + n] = -acc[jt][r];
}

extern "C" void kernel_launch(void* const* d_in, const int* in_sizes, int n_in,
                              void* d_out, int out_size, void* d_ws, size_t ws_size,
                              hipStream_t stream) {
    (void)n_in; (void)out_size; (void)d_ws; (void)ws_size;
    const float* muTE  = (const float*)d_in[0];
    const float* covTE = (const float*)d_in[1];
    const float* chol  = (const float*)d_in[2];
    float* out = (float*)d_out;

    const int nbatch = in_sizes[0] / NN;                 // B = 4096
    const size_t shmem = (size_t)2 * NN * LS * sizeof(float);   // ~132 KB

    hipFuncSetAttribute((const void*)gauss_te_kernel,
                        hipFuncAttributeMaxDynamicSharedMemorySize, (int)shmem);
    gauss_te_kernel<<<nbatch, 256, shmem, stream>>>(muTE, covTE, chol, out, nbatch);
}